// EQGraphNet_54065048322409
// MI455X (gfx1250) — compile-verified
//
#include <hip/hip_runtime.h>

// ---------------------------------------------------------------------------
// EQGraphNet on MI455X (gfx1250).
//  * fp32 WMMA (V_WMMA_F32_16X16X4_F32) for conv-as-GEMM and the GCN layer.
//  * GCN layer fully fused: agg(W h) == W agg(h) (aggregation is a scalar-
//    weighted node sum, W acts on channels only), so one kernel builds the
//    band-aggregated B-fragment on the fly and computes linear + bias +
//    residual in a single pass. No `ht` intermediate; 2 ping-pong buffers
//    (~196 MB, close to the 192 MB L2) are the whole working set.
//  * Workload is HBM-bandwidth shaped (~6 GFLOP, few hundred MB traffic at
//    23.3 TB/s), so fp32 matrix pipe + minimal passes is the optimum; lower
//    precision buys nothing.
//  * Fixed vs previous round: buffer rotation bug that aliased p0/p1 after
//    stage 0 (gcn would have read and written the same buffer). Scheme is
//    now: conv: cur -> p0 ; gcn: p0 -> p1 ; cur = p1. Stream ordering makes
//    overwriting the conv's already-consumed input safe.
// ---------------------------------------------------------------------------

typedef float v2f __attribute__((ext_vector_type(2)));
typedef float v8f __attribute__((ext_vector_type(8)));

#define NBATCH 512

// ---------------------------------------------------------------------------
// Conv1d (kernel=2, stride=2, VALID) as GEMM:
//   Y[b, co, t] = bias[co] + sum_{ci,kk} W[co, ci*2+kk] * maybe_relu(X[b, ci, 2t+kk])
// One wave computes a 16(M=co) x 16(col=b*Nout+t) tile.
// Fragment layout (ISA 7.12.2):
//   A 16x4 : lanes 0-15 -> {K=k0,k0+1}, lanes 16-31 -> {K=k0+2,k0+3}, M = lane&15
//   B 4x16 : same K split, N = lane&15
//   C/D    : vgpr r: M = mt*16 + r + (lane>=16 ? 8 : 0), N = lane&15
// ---------------------------------------------------------------------------
__global__ __launch_bounds__(256) void conv_wmma_kernel(
    const float* __restrict__ in, const float* __restrict__ w,
    const float* __restrict__ bias, float* __restrict__ out,
    int Cin, int Cout, int Nin, int Nout, int relu_in, int tiles)
{
  const int wave = (blockIdx.x << 3) + (threadIdx.x >> 5);
  if (wave >= tiles) return;                       // wave-uniform branch
  const int lane = threadIdx.x & 31;
  const int l16  = lane & 15;
  const int hi   = lane >> 4;

  const int Mtiles = Cout >> 4;
  const int mt = wave % Mtiles;
  const int nt = wave / Mtiles;
  const int K  = Cin * 2;

  const int col = (nt << 4) + l16;                 // flat output column
  const int b   = col / Nout;
  const int t   = col - b * Nout;
  const int inB = b * Cin * Nin + 2 * t;           // base of (b, ci=0, 2t)
  const int mrow = (mt << 4) + l16;                // A-matrix row

  v8f acc = {0.f, 0.f, 0.f, 0.f, 0.f, 0.f, 0.f, 0.f};

  for (int k0 = 0; k0 < K; k0 += 4) {
    const int kb = k0 + (hi << 1);
    v2f a, bfrag;
    a.x = (kb     < K) ? w[mrow * K + kb    ] : 0.f;
    a.y = (kb + 1 < K) ? w[mrow * K + kb + 1] : 0.f;
    float v0 = 0.f, v1 = 0.f;
    if (kb     < K) v0 = in[inB + ((kb    ) >> 1) * Nin + ((kb    ) & 1)];
    if (kb + 1 < K) v1 = in[inB + ((kb + 1) >> 1) * Nin + ((kb + 1) & 1)];
    if (relu_in) { v0 = fmaxf(v0, 0.f); v1 = fmaxf(v1, 0.f); }
    bfrag.x = v0; bfrag.y = v1;
    acc = __builtin_amdgcn_wmma_f32_16x16x4_f32(
        false, a, false, bfrag, (short)0, acc, false, false);
  }

  const int rbase = (mt << 4) + (hi << 3);
  const int obase = (b * Cout + rbase) * Nout + t;
#pragma unroll
  for (int r = 0; r < 8; ++r)
    out[obase + r * Nout] = acc[r] + bias[rbase + r];
}

// ---------------------------------------------------------------------------
// Fused GCN layer. Uses agg(W h) == W agg(h):
//   g[b,c,n]   = sum_{d=-3..3, 0<=n+d<N} deg[n+d]^-.5 deg[n]^-.5 * h0[b,c,n+d]
//   out[b,d,n] = h0[b,d,n] + bias[d] + sum_c W[d,c] * g[b,c,n]
// deg[i] = min(i,3)+min(N-1-i,3)+1 reproduces _build_edges exactly.
// One wave owns one 16-column n-tile and computes ALL MT=C/16 row tiles,
// so each on-the-fly g fragment is reused MT times. Out-of-range neighbors
// use clamped addresses with zero weights (no OOB access, batch-safe).
// ---------------------------------------------------------------------------
template <int MT>
__global__ __launch_bounds__(256) void gcn_fused_kernel(
    const float* __restrict__ h0, const float* __restrict__ w,
    const float* __restrict__ bias, float* __restrict__ out,
    int N, int ntiles)
{
  constexpr int C = MT * 16;
  const int wave = (blockIdx.x << 3) + (threadIdx.x >> 5);
  if (wave >= ntiles) return;                      // wave-uniform branch
  const int lane = threadIdx.x & 31;
  const int l16  = lane & 15;
  const int hi   = lane >> 4;

  const int col = (wave << 4) + l16;
  const int b   = col / N;
  const int t   = col - b * N;

  // per-lane band weights + clamped neighbor offsets
  const float rn = rsqrtf((float)(min(t, 3) + min(N - 1 - t, 3) + 1));
  float wgt[7];
  int   moff[7];
#pragma unroll
  for (int d = 0; d < 7; ++d) {
    const int  m  = t + d - 3;
    const bool v  = (m >= 0) && (m < N);
    const int  mc = min(max(m, 0), N - 1);
    moff[d] = mc;
    wgt[d]  = v ? rsqrtf((float)(min(mc, 3) + min(N - 1 - mc, 3) + 1)) * rn : 0.f;
  }

  const int bbase = b * C * N;

  v8f acc[MT];
#pragma unroll
  for (int m = 0; m < MT; ++m)
    acc[m] = (v8f){0.f, 0.f, 0.f, 0.f, 0.f, 0.f, 0.f, 0.f};

#pragma unroll
  for (int k0 = 0; k0 < C; k0 += 4) {
    const int kb = k0 + (hi << 1);
    const float* __restrict__ r0 = h0 + bbase + kb * N;
    const float* __restrict__ r1 = r0 + N;
    float g0 = 0.f, g1 = 0.f;
#pragma unroll
    for (int d = 0; d < 7; ++d) {
      g0 += wgt[d] * r0[moff[d]];
      g1 += wgt[d] * r1[moff[d]];
    }
    v2f bfrag; bfrag.x = g0; bfrag.y = g1;
#pragma unroll
    for (int m = 0; m < MT; ++m) {
      v2f a;
      a.x = w[(m * 16 + l16) * C + kb    ];
      a.y = w[(m * 16 + l16) * C + kb + 1];
      acc[m] = __builtin_amdgcn_wmma_f32_16x16x4_f32(
          false, a, false, bfrag, (short)0, acc[m], false, false);
    }
  }

#pragma unroll
  for (int m = 0; m < MT; ++m) {
    const int rb = m * 16 + (hi << 3);
    const int ob = bbase + rb * N + t;
#pragma unroll
    for (int r = 0; r < 8; ++r)
      out[ob + r * N] = acc[m][r] + bias[rb + r] + h0[ob + r * N];
  }
}

// ---------------------------------------------------------------------------
// Final linear: out[b] = dot(h[b, 0:256], lin_w) + lin_b   (h = [512,128,2])
// One wave per batch element, wave32 shuffle reduction.
// ---------------------------------------------------------------------------
__global__ __launch_bounds__(256) void final_linear_kernel(
    const float* __restrict__ h, const float* __restrict__ w,
    const float* __restrict__ lb, float* __restrict__ out)
{
  const int b    = blockIdx.x * 8 + (threadIdx.x >> 5);
  const int lane = threadIdx.x & 31;
  float s = 0.f;
#pragma unroll
  for (int j = lane; j < 256; j += 32) s += h[b * 256 + j] * w[j];
#pragma unroll
  for (int off = 16; off > 0; off >>= 1) s += __shfl_down(s, off, 32);
  if (lane == 0) out[b] = s + lb[0];
}

// ---------------------------------------------------------------------------
// Host orchestration
// ---------------------------------------------------------------------------
static const int CIN [11] = {3, 16, 16, 16, 32, 32, 32, 64, 64, 64, 128};
static const int COUT[11] = {16, 16, 16, 32, 32, 32, 64, 64, 64, 128, 128};
static const int NIN [11] = {6000, 3000, 1500, 750, 375, 187, 93, 46, 23, 11, 5};
static const int NOUT[11] = {3000, 1500, 750, 375, 187, 93, 46, 23, 11, 5, 2};

// Max activation footprint: 512 * 16 * 3000 floats = 98.3 MB per buffer.
#define MAXELEM (512 * 16 * 3000)

static void launch_gcn(const float* h0, const float* w, const float* bias,
                       float* out, int C, int N, hipStream_t stream) {
  const int ntiles = (NBATCH * N) >> 4;            // 32*N, multiple of 32
  const int grid   = ntiles >> 3;
  switch (C >> 4) {
    case 1: gcn_fused_kernel<1><<<grid, 256, 0, stream>>>(h0, w, bias, out, N, ntiles); break;
    case 2: gcn_fused_kernel<2><<<grid, 256, 0, stream>>>(h0, w, bias, out, N, ntiles); break;
    case 4: gcn_fused_kernel<4><<<grid, 256, 0, stream>>>(h0, w, bias, out, N, ntiles); break;
    default: gcn_fused_kernel<8><<<grid, 256, 0, stream>>>(h0, w, bias, out, N, ntiles); break;
  }
}

extern "C" void kernel_launch(void* const* d_in, const int* in_sizes, int n_in,
                              void* d_out, int out_size, void* d_ws, size_t ws_size,
                              hipStream_t stream) {
  (void)in_sizes; (void)n_in; (void)out_size; (void)ws_size;

  const float* x = (const float*)d_in[0];
  const float* conv_w[11];
  const float* conv_b[11];
  const float* gnn_w[10];
  const float* gnn_b[10];
  for (int i = 0; i < 11; ++i) conv_w[i] = (const float*)d_in[1 + i];
  for (int i = 0; i < 11; ++i) conv_b[i] = (const float*)d_in[12 + i];
  for (int i = 0; i < 10; ++i) gnn_w[i]  = (const float*)d_in[23 + i];
  for (int i = 0; i < 10; ++i) gnn_b[i]  = (const float*)d_in[33 + i];
  const float* lin_w = (const float*)d_in[43];
  const float* lin_b = (const float*)d_in[44];
  // edge_srcs/dsts/norms (d_in[45..74]) are reproduced analytically.

  float* p0 = (float*)d_ws;                        // conv output buffer
  float* p1 = p0 + MAXELEM;                        // gcn output buffer

  // Fixed schedule (no aliasing):
  //   stage i: conv: cur -> p0 ; gcn: p0 -> p1 ; cur = p1
  // p0 is always free when conv runs (previous gcn already consumed it);
  // p1 (== cur) is free when gcn runs (conv already consumed it; kernels are
  // stream-ordered).
  const float* cur = x;
  for (int i = 0; i < 10; ++i) {
    const int Cc = COUT[i], Nn = NOUT[i];
    // conv i (ReLU on input for i > 0): cur -> p0
    {
      const int tiles = (Cc >> 4) * ((NBATCH * Nn) >> 4);  // multiple of 32
      conv_wmma_kernel<<<tiles >> 3, 256, 0, stream>>>(
          cur, conv_w[i], conv_b[i], p0,
          CIN[i], Cc, NIN[i], Nn, (i > 0) ? 1 : 0, tiles);
    }
    // fused GCN (linear + band-aggregate + bias + residual): p0 -> p1
    launch_gcn(p0, gnn_w[i], gnn_b[i], p1, Cc, Nn, stream);
    cur = p1;
  }

  // conv10: [512,128,5] -> [512,128,2], ReLU on input: cur(p1) -> p0
  {
    const int tiles = (COUT[10] >> 4) * ((NBATCH * NOUT[10]) >> 4);  // 512
    conv_wmma_kernel<<<tiles >> 3, 256, 0, stream>>>(
        cur, conv_w[10], conv_b[10], p0,
        CIN[10], COUT[10], NIN[10], NOUT[10], 1, tiles);
  }

  // final linear -> d_out[512]
  final_linear_kernel<<<NBATCH / 8, 256, 0, stream>>>(
      p0, lin_w, lin_b, (float*)d_out);
}